// GraphTimeSeriesEmbedding_51427938402412
// MI455X (gfx1250) — compile-verified
//
#include <hip/hip_runtime.h>

typedef __attribute__((ext_vector_type(2))) float v2f;
typedef __attribute__((ext_vector_type(8))) float v8f;

#define D_MODEL 128
#define C_DIM   32
#define N_NODES 64
#define L_SEQ   512
// M = B*L*N = 16*512*64 = 524288 rows; 128 rows per block -> 4096 blocks
#define ROWS_PER_BLOCK 128

__global__ __launch_bounds__(256)
void graph_ts_embed_wmma(const float* __restrict__ x,
                         const float* __restrict__ W,
                         const float* __restrict__ bias,
                         const float* __restrict__ spatial,
                         float* __restrict__ out)
{
    __shared__ float lds_W[C_DIM * D_MODEL];     // 16 KB, K-major: lds_W[k*128 + d]
    __shared__ float lds_sp[N_NODES * D_MODEL];  // 32 KB
    __shared__ float lds_b[D_MODEL];

    const int tid = threadIdx.x;

    // ---- cooperative staging (coalesced b128 loads; tables are L2-resident) ----
    {
        const float4* src = (const float4*)W;
        float4*       dst = (float4*)lds_W;
        #pragma unroll
        for (int i = 0; i < (C_DIM * D_MODEL / 4) / 256; ++i)
            dst[tid + i * 256] = src[tid + i * 256];
    }
    {
        const float4* src = (const float4*)spatial;
        float4*       dst = (float4*)lds_sp;
        #pragma unroll
        for (int i = 0; i < (N_NODES * D_MODEL / 4) / 256; ++i)
            dst[tid + i * 256] = src[tid + i * 256];
    }
    if (tid < D_MODEL) lds_b[tid] = bias[tid];
    __syncthreads();

    const int wave = tid >> 5;   // 0..7
    const int lane = tid & 31;
    const int lo   = lane & 15;
    const int hi   = lane >> 4;

    const long long m0 = (long long)blockIdx.x * ROWS_PER_BLOCK + (long long)wave * 16;
    const int l  = (int)((m0 >> 6) & (L_SEQ - 1));   // sequence position (constant per strip)
    const int n0 = (int)(m0 & (N_NODES - 1));        // node base (constant per strip)

    // ---- A fragments: x[m0..m0+15, 0..31], 8 K-chunks of 4 ----
    // ISA 32-bit 16x4 A layout: lanes 0-15 hold M=lane, VGPR0=K0,VGPR1=K1;
    // lanes 16-31 hold M=lane-16, VGPR0=K2,VGPR1=K3.
    v2f a[8];
    {
        const float* xrow = x + (m0 + lo) * C_DIM + 2 * hi;
        #pragma unroll
        for (int kc = 0; kc < 8; ++kc) {
            // x is streamed read-once: non-temporal loads keep L2 for the tables
            a[kc].x = __builtin_nontemporal_load(&xrow[4 * kc + 0]);
            a[kc].y = __builtin_nontemporal_load(&xrow[4 * kc + 1]);
        }
    }

    const float NEG_2LN1E4_OVER_D = -2.0f * 9.210340372f / (float)D_MODEL; // -2*ln(10000)/128
    const float HALF_PI = 1.5707963267948966f;

    // ---- 8 column tiles of 16 ----
    #pragma unroll 1
    for (int nt = 0; nt < 8; ++nt) {
        const int colg = nt * 16 + lo;

        // B fragments from LDS (4x16 f32 per chunk, layout mirrors A)
        v8f acc = {0.f, 0.f, 0.f, 0.f, 0.f, 0.f, 0.f, 0.f};
        #pragma unroll
        for (int kc = 0; kc < 8; ++kc) {
            v2f bf;
            bf.x = lds_W[(4 * kc + 2 * hi + 0) * D_MODEL + colg];
            bf.y = lds_W[(4 * kc + 2 * hi + 1) * D_MODEL + colg];
            acc = __builtin_amdgcn_wmma_f32_16x16x4_f32(
                      /*neg_a=*/false, a[kc],
                      /*neg_b=*/false, bf,
                      /*c_mod=*/(short)0, acc,
                      /*reuse_a=*/false, /*reuse_b=*/false);
        }

        // ---- epilogue: 2*acc + 2*b[col] + pe[l,col] + spatial[n0+row,col] ----
        // C/D layout: VGPR i -> row (i + 8*hi), col = lane&15 (fixed per lane).
        // pe: even cols sin(l*dt), odd cols cos(l*dt) == sin(l*dt + pi/2) -> branch-free
        const int   j    = colg >> 1;
        const float ang  = (float)l * __expf((float)j * NEG_2LN1E4_OVER_D)
                         + (float)(colg & 1) * HALF_PI;
        const float pe   = __sinf(ang);
        const float addc = 2.0f * lds_b[colg] + pe;

        float* orow = out + m0 * D_MODEL + colg;
        #pragma unroll
        for (int i = 0; i < 8; ++i) {
            const int r = i + 8 * hi;
            const float v = 2.0f * acc[i] + addc + lds_sp[(n0 + r) * D_MODEL + colg];
            // out is write-once streaming: non-temporal store
            __builtin_nontemporal_store(v, &orow[(long long)r * D_MODEL]);
        }
    }
}

extern "C" void kernel_launch(void* const* d_in, const int* in_sizes, int n_in,
                              void* d_out, int out_size, void* d_ws, size_t ws_size,
                              hipStream_t stream) {
    (void)in_sizes; (void)n_in; (void)d_ws; (void)ws_size; (void)out_size;
    const float* x       = (const float*)d_in[0];  // [16,512,64,32]
    const float* W       = (const float*)d_in[1];  // [32,128]
    const float* bias    = (const float*)d_in[2];  // [128]
    const float* spatial = (const float*)d_in[3];  // [64,128]
    float*       out     = (float*)d_out;          // [16,512,64,128]

    const long long M = 16LL * 512LL * 64LL;       // 524288 rows
    const int blocks = (int)(M / ROWS_PER_BLOCK);  // 4096
    graph_ts_embed_wmma<<<blocks, 256, 0, stream>>>(x, W, bias, spatial, out);
}